// GAT_48464410968743
// MI455X (gfx1250) — compile-verified
//
#include <hip/hip_runtime.h>
#include <hip/hip_bf16.h>

// ---------------------------------------------------------------------------
// GAT-style op on MI455X (gfx1250), fp32 throughout via V_WMMA_F32_16X16X4_F32.
// N = M = 4096, D_MODEL = 128, H = 6, D = 32, H*D = 192.
// ---------------------------------------------------------------------------

typedef __attribute__((ext_vector_type(2))) float v2f;
typedef __attribute__((ext_vector_type(8))) float v8f;

#define NNODES 4096
#define DMODEL 128
#define NHEAD  6
#define HD     192
#define DH     32
#define ALPHA  0.2f
#define NEGBIG -9e15f
#define ROWCHUNK 256

__device__ __forceinline__ v8f wmma4(v2f a, v2f b, v8f c) {
    // D = A(16x4,f32) * B(4x16,f32) + C(16x16,f32)
    return __builtin_amdgcn_wmma_f32_16x16x4_f32(
        /*neg_a=*/false, a, /*neg_b=*/false, b,
        /*c_mod=*/(short)0, c, /*reuse_a=*/false, /*reuse_b=*/false);
}

__device__ __forceinline__ float waveSum(float v) {
    #pragma unroll
    for (int o = 16; o; o >>= 1) v += __shfl_xor(v, o, 32);
    return v;
}
__device__ __forceinline__ float waveMax(float v) {
    #pragma unroll
    for (int o = 16; o; o >>= 1) v = fmaxf(v, __shfl_xor(v, o, 32));
    return v;
}
__device__ __forceinline__ float blockSum(float v, float* red) {
    int lane = threadIdx.x & 31, wid = threadIdx.x >> 5;
    int nw = (blockDim.x + 31) >> 5;
    v = waveSum(v);
    __syncthreads();
    if (lane == 0) red[wid] = v;
    __syncthreads();
    float r = (lane < nw) ? red[lane] : 0.0f;
    return waveSum(r);
}
__device__ __forceinline__ float blockMax(float v, float* red) {
    int lane = threadIdx.x & 31, wid = threadIdx.x >> 5;
    int nw = (blockDim.x + 31) >> 5;
    v = waveMax(v);
    __syncthreads();
    if (lane == 0) red[wid] = v;
    __syncthreads();
    float r = (lane < nw) ? red[lane] : -3.4e38f;
    return waveMax(r);
}

// ---------------------------------------------------------------------------
// 1) Projection: out[4096,192] = X[4096,128] @ W[192,128]^T + b[192]
// ---------------------------------------------------------------------------
__global__ void proj_kernel(const float* __restrict__ X,
                            const float* __restrict__ W,
                            const float* __restrict__ bias,
                            float* __restrict__ out) {
    int lane = threadIdx.x & 31;
    int half = lane >> 4, q = lane & 15;
    int r0 = blockIdx.x * 16;           // 256 row tiles
    int c0 = blockIdx.y * 16;           // 12 col tiles
    const float* ap = X + (size_t)(r0 + q) * DMODEL + half * 2;
    const float* bp = W + (size_t)(c0 + q) * DMODEL + half * 2;
    v8f c = {};
    #pragma unroll
    for (int k = 0; k < DMODEL; k += 4) {
        v2f a = {ap[k], ap[k + 1]};
        v2f b = {bp[k], bp[k + 1]};
        c = wmma4(a, b, c);
    }
    float bv = bias[c0 + q];
    int orow = r0 + 8 * half;
    #pragma unroll
    for (int i = 0; i < 8; i++)
        out[(size_t)(orow + i) * HD + c0 + q] = c[i] + bv;
}

// ---------------------------------------------------------------------------
// 2) Scores: e[n,m] = mask( (1/H) * sum_h lrelu( sum_d Xh[n,d]*wx[d]*Xh[m,d] ) )
//    P viewed as [H*4096, 32] row-major == Xh flat. One wave per 16x16 tile.
// ---------------------------------------------------------------------------
__global__ void scores_kernel(const float* __restrict__ P,
                              const float* __restrict__ wx,
                              const int* __restrict__ adj,
                              float* __restrict__ e) {
    int wave = threadIdx.x >> 5;
    int lane = threadIdx.x & 31;
    int half = lane >> 4, q = lane & 15;
    int n0 = blockIdx.x * 16;
    int m0 = (blockIdx.y * 8 + wave) * 16;
    v8f acc = {};
    for (int h = 0; h < NHEAD; h++) {
        const float* Ah = P + (size_t)(h * NNODES + n0 + q) * DH + half * 2;
        const float* Bh = P + (size_t)(h * NNODES + m0 + q) * DH + half * 2;
        v8f c = {};
        #pragma unroll
        for (int k = 0; k < DH; k += 4) {
            float w0 = wx[k + half * 2];
            float w1 = wx[k + half * 2 + 1];
            v2f a = {Ah[k] * w0, Ah[k + 1] * w1};
            v2f b = {Bh[k], Bh[k + 1]};
            c = wmma4(a, b, c);
        }
        #pragma unroll
        for (int i = 0; i < 8; i++) {
            float v = c[i];
            acc[i] += (v > 0.0f) ? v : ALPHA * v;
        }
    }
    int row = n0 + 8 * half;
    int col = m0 + q;
    #pragma unroll
    for (int i = 0; i < 8; i++) {
        float v = acc[i] * (1.0f / (float)NHEAD);
        int a = adj[(size_t)(row + i) * NNODES + col];
        e[(size_t)(row + i) * NNODES + col] = (a > 0) ? v : NEGBIG;
    }
}

// ---------------------------------------------------------------------------
// 3) Column softmax (axis 0), split for parallelism. All sweeps coalesced
//    (consecutive threads = consecutive columns = consecutive addresses).
//    Normalization (1/colsum) is NOT applied here; it is folded into the
//    A-fragments of the xprime GEMM (m == WMMA K index).
// ---------------------------------------------------------------------------
__global__ void colmax_part_kernel(const float* __restrict__ E,
                                   float* __restrict__ pmax) {
    int m = blockIdx.x * 256 + threadIdx.x;
    int r0 = blockIdx.y * ROWCHUNK;
    float mx = -3.4e38f;
    for (int n = r0; n < r0 + ROWCHUNK; n++)
        mx = fmaxf(mx, E[(size_t)n * NNODES + m]);
    pmax[(size_t)blockIdx.y * NNODES + m] = mx;
}
__global__ void colmax_reduce_kernel(const float* __restrict__ pmax,
                                     float* __restrict__ cmax) {
    int m = blockIdx.x * 256 + threadIdx.x;
    float mx = -3.4e38f;
    #pragma unroll
    for (int j = 0; j < NNODES / ROWCHUNK; j++)
        mx = fmaxf(mx, pmax[(size_t)j * NNODES + m]);
    cmax[m] = mx;
}
__global__ void colexp_part_kernel(float* __restrict__ E,
                                   const float* __restrict__ cmax,
                                   float* __restrict__ psum) {
    int m = blockIdx.x * 256 + threadIdx.x;
    int r0 = blockIdx.y * ROWCHUNK;
    float mx = cmax[m];
    float s = 0.0f;
    for (int n = r0; n < r0 + ROWCHUNK; n++) {
        float v = __expf(E[(size_t)n * NNODES + m] - mx);
        E[(size_t)n * NNODES + m] = v;
        s += v;
    }
    psum[(size_t)blockIdx.y * NNODES + m] = s;
}
__global__ void colsum_reduce_kernel(const float* __restrict__ psum,
                                     float* __restrict__ inv) {
    int m = blockIdx.x * 256 + threadIdx.x;
    float s = 0.0f;
    #pragma unroll
    for (int j = 0; j < NNODES / ROWCHUNK; j++)
        s += psum[(size_t)j * NNODES + m];
    inv[m] = 1.0f / s;
}

// ---------------------------------------------------------------------------
// 4) e_xy[n] = (1/H) sum_h sum_m lrelu( sum_d Xh[n,d]*wxy[d]*Yh[m,d] )
// ---------------------------------------------------------------------------
__global__ void exy_kernel(const float* __restrict__ P,
                           const float* __restrict__ Q,
                           const float* __restrict__ wxy,
                           float* __restrict__ exy) {
    int wave = threadIdx.x >> 5;
    int lane = threadIdx.x & 31;
    int half = lane >> 4, q = lane & 15;
    int n0 = blockIdx.x * 16;
    int m0 = (blockIdx.y * 8 + wave) * 16;
    v8f acc = {};
    for (int h = 0; h < NHEAD; h++) {
        const float* Ah = P + (size_t)(h * NNODES + n0 + q) * DH + half * 2;
        const float* Bh = Q + (size_t)(h * NNODES + m0 + q) * DH + half * 2;
        v8f c = {};
        #pragma unroll
        for (int k = 0; k < DH; k += 4) {
            float w0 = wxy[k + half * 2];
            float w1 = wxy[k + half * 2 + 1];
            v2f a = {Ah[k] * w0, Ah[k + 1] * w1};
            v2f b = {Bh[k], Bh[k + 1]};
            c = wmma4(a, b, c);
        }
        #pragma unroll
        for (int i = 0; i < 8; i++) {
            float v = c[i];
            acc[i] += (v > 0.0f) ? v : ALPHA * v;
        }
    }
    // sum over the 16 columns held by each half-wave, add into exy[row]
    #pragma unroll
    for (int i = 0; i < 8; i++) {
        float v = acc[i] * (1.0f / (float)NHEAD);
        #pragma unroll
        for (int o = 1; o < 16; o <<= 1) v += __shfl_xor(v, o, 32);
        if (q == 0) atomicAdd(&exy[n0 + i + 8 * half], v);
    }
}

// ---------------------------------------------------------------------------
// 5) att_xy = softmax(e_xy)  (single block, 1024 threads, 4 elems/thread)
// ---------------------------------------------------------------------------
__global__ void vec_softmax_kernel(const float* __restrict__ in,
                                   float* __restrict__ out) {
    __shared__ float red[32];
    int t = threadIdx.x;
    float v[4];
    float mx = -3.4e38f;
    #pragma unroll
    for (int j = 0; j < 4; j++) { v[j] = in[t + j * 1024]; mx = fmaxf(mx, v[j]); }
    mx = blockMax(mx, red);
    float se = 0.0f;
    #pragma unroll
    for (int j = 0; j < 4; j++) { v[j] = __expf(v[j] - mx); se += v[j]; }
    se = blockSum(se, red);
    float inv = 1.0f / se;
    #pragma unroll
    for (int j = 0; j < 4; j++) out[t + j * 1024] = v[j] * inv;
}

// ---------------------------------------------------------------------------
// 6) x_prime[h,n,d] = sum_m (E[n,m]*inv[m]) * Xh[h,m,d]   (NN GEMM, K=4096)
//    One wave per (n-tile, d-tile) computes ALL 6 heads, reusing each
//    A-fragment 6x (cuts att traffic 6x). inv[] normalization folded into A.
// ---------------------------------------------------------------------------
__global__ void xprime_kernel(const float* __restrict__ E,
                              const float* __restrict__ inv,
                              const float* __restrict__ P,
                              float* __restrict__ xp) {
    int wave = threadIdx.x >> 5;
    int lane = threadIdx.x & 31;
    int half = lane >> 4, q = lane & 15;
    int n0 = (blockIdx.x * 4 + (wave >> 1)) * 16;   // 64 blocks * 4 = 256 n-tiles
    int d0 = (wave & 1) * 16;                       // 2 d-tiles per n-tile
    const float* Arow = E + (size_t)(n0 + q) * NNODES + half * 2;
    const float* Bp   = P + d0 + q;  // B[k][col] = P[(h*4096+k)*32 + d0+col]
    v8f c[NHEAD];
    #pragma unroll
    for (int h = 0; h < NHEAD; h++) c[h] = (v8f){};
    #pragma unroll 4
    for (int k = 0; k < NNODES; k += 4) {
        int kk = k + half * 2;
        v2f a = {Arow[k] * inv[kk], Arow[k + 1] * inv[kk + 1]};
        #pragma unroll
        for (int h = 0; h < NHEAD; h++) {
            const float* bp = Bp + (size_t)(h * NNODES + kk) * DH;
            v2f b = {bp[0], bp[DH]};
            c[h] = wmma4(a, b, c[h]);
        }
    }
    int orow = n0 + 8 * half;
    #pragma unroll
    for (int h = 0; h < NHEAD; h++)
        #pragma unroll
        for (int i = 0; i < 8; i++)
            xp[(size_t)h * NNODES * DH + (size_t)(orow + i) * DH + d0 + q] = c[h][i];
}

// ---------------------------------------------------------------------------
// 7) s[r] = sum over 192 consecutive flat elems of z, where
//    z_flat[i] = xp_flat[i] + 0.5 * att_xy[(i>>5)&4095] * P_flat[i]
//    (reproduces torch's .view(N,-1).sum(1) exactly)
// ---------------------------------------------------------------------------
__global__ void srow_kernel(const float* __restrict__ xp,
                            const float* __restrict__ P,
                            const float* __restrict__ axy,
                            float* __restrict__ s) {
    int r = blockIdx.x;
    int lane = threadIdx.x;
    size_t base = (size_t)r * HD;
    float acc = 0.0f;
    for (int j = lane; j < HD; j += 32) {
        size_t i = base + j;
        int n = (int)((i >> 5) & (NNODES - 1));
        acc += xp[i] + 0.5f * axy[n] * P[i];
    }
    acc = waveSum(acc);
    if (lane == 0) s[r] = acc;
}

// ---------------------------------------------------------------------------
// 8) LayerNorm(4096-vector) -> sigmoid -> softmax  (single block)
// ---------------------------------------------------------------------------
__global__ void finish_kernel(const float* __restrict__ s,
                              const float* __restrict__ lnw,
                              const float* __restrict__ lnb,
                              float* __restrict__ out) {
    __shared__ float red[32];
    int t = threadIdx.x;
    float v[4];
    float sum = 0.0f;
    #pragma unroll
    for (int j = 0; j < 4; j++) { v[j] = s[t + j * 1024]; sum += v[j]; }
    sum = blockSum(sum, red);
    float mu = sum * (1.0f / (float)NNODES);
    float vr = 0.0f;
    #pragma unroll
    for (int j = 0; j < 4; j++) { float d = v[j] - mu; vr += d * d; }
    vr = blockSum(vr, red);
    float rstd = rsqrtf(vr * (1.0f / (float)NNODES) + 1e-5f);
    float mm[4];
    #pragma unroll
    for (int j = 0; j < 4; j++) {
        int idx = t + j * 1024;
        float x = (v[j] - mu) * rstd * lnw[idx] + lnb[idx];
        mm[j] = 1.0f / (1.0f + __expf(-x));
    }
    float mx = -3.4e38f;
    #pragma unroll
    for (int j = 0; j < 4; j++) mx = fmaxf(mx, mm[j]);
    mx = blockMax(mx, red);
    float se = 0.0f;
    #pragma unroll
    for (int j = 0; j < 4; j++) { mm[j] = __expf(mm[j] - mx); se += mm[j]; }
    se = blockSum(se, red);
    float inv = 1.0f / se;
    #pragma unroll
    for (int j = 0; j < 4; j++) out[t + j * 1024] = mm[j] * inv;
}

__global__ void zero_kernel(float* __restrict__ p, int n) {
    int i = blockIdx.x * blockDim.x + threadIdx.x;
    if (i < n) p[i] = 0.0f;
}

// ---------------------------------------------------------------------------
extern "C" void kernel_launch(void* const* d_in, const int* in_sizes, int n_in,
                              void* d_out, int out_size, void* d_ws, size_t ws_size,
                              hipStream_t stream) {
    (void)in_sizes; (void)n_in; (void)out_size; (void)ws_size;

    const float* input_x = (const float*)d_in[0];   // [4096,128]
    const float* input_y = (const float*)d_in[1];   // [4096,128]
    const float* Wx      = (const float*)d_in[2];   // [192,128]
    const float* bx      = (const float*)d_in[3];   // [192]
    const float* Wy      = (const float*)d_in[4];   // [192,128]
    const float* by      = (const float*)d_in[5];   // [192]
    const float* wx      = (const float*)d_in[6];   // [32]
    const float* wxy     = (const float*)d_in[7];   // [32]
    const float* ln_w    = (const float*)d_in[8];   // [4096]
    const float* ln_b    = (const float*)d_in[9];   // [4096]
    const int*   adj     = (const int*)d_in[10];    // [4096,4096]
    float* out = (float*)d_out;                     // [4096]

    // workspace layout (floats); total ~77.5 MB
    float* ws   = (float*)d_ws;
    float* P    = ws;                  // 786432  : X projection == Xh flat
    float* Q    = P + 786432;          // 786432  : Y projection == Yh flat
    float* XP   = Q + 786432;          // 786432  : x_prime flat
    float* EXY  = XP + 786432;         // 4096
    float* AXY  = EXY + 4096;          // 4096
    float* S    = AXY + 4096;          // 4096
    float* CMAX = S + 4096;            // 4096    : per-column max
    float* CINV = CMAX + 4096;         // 4096    : 1/colsum
    float* PMAX = CINV + 4096;         // 65536   : partial col maxes [16][4096]
    float* PSUM = PMAX + 65536;        // 65536   : partial col sums  [16][4096]
    float* E    = PSUM + 65536;        // 16777216: scores -> exp(scores) in place

    // 1) projections
    proj_kernel<<<dim3(256, 12), 32, 0, stream>>>(input_x, Wx, bx, P);
    proj_kernel<<<dim3(256, 12), 32, 0, stream>>>(input_y, Wy, by, Q);

    // 2) masked scores e[4096,4096]
    scores_kernel<<<dim3(256, 32), 256, 0, stream>>>(P, wx, adj, E);

    // 3) column softmax statistics (exp in place; normalization folded into 6)
    colmax_part_kernel<<<dim3(16, 16), 256, 0, stream>>>(E, PMAX);
    colmax_reduce_kernel<<<16, 256, 0, stream>>>(PMAX, CMAX);
    colexp_part_kernel<<<dim3(16, 16), 256, 0, stream>>>(E, CMAX, PSUM);
    colsum_reduce_kernel<<<16, 256, 0, stream>>>(PSUM, CINV);

    // 4) e_xy
    zero_kernel<<<16, 256, 0, stream>>>(EXY, 4096);
    exy_kernel<<<dim3(256, 32), 256, 0, stream>>>(P, Q, wxy, EXY);

    // 5) att_xy
    vec_softmax_kernel<<<1, 1024, 0, stream>>>(EXY, AXY);

    // 6) x_prime = att @ Xh, all heads per wave, normalization fused
    xprime_kernel<<<dim3(64, 1), 256, 0, stream>>>(E, CINV, P, XP);

    // 7) per-row z sums
    srow_kernel<<<4096, 32, 0, stream>>>(XP, P, AXY, S);

    // 8) LayerNorm + sigmoid + softmax
    finish_kernel<<<1, 1024, 0, stream>>>(S, ln_w, ln_b, out);
}